// ImageCaptionModel_14671608283463
// MI455X (gfx1250) — compile-verified
//
#include <hip/hip_runtime.h>
#include <math.h>

// ---------------------------------------------------------------------------
// ImageCaptionModel on gfx1250 (CDNA5) using V_WMMA_F32_16X16X4_F32.
// Key exact simplification: softmax over a length-1 axis == 1.0, so the
// attention branch is dead and context == img_out for every timestep.
// ---------------------------------------------------------------------------

#define B_  128
#define T_  32
#define H_  1024
#define E_  512
#define F_  2048
#define V_  32000

typedef __attribute__((ext_vector_type(2))) float v2f;
typedef __attribute__((ext_vector_type(8))) float v8f;

// ---------------------------------------------------------------------------
// rowidx[m] for m = t*B + b  ->  seq[b][t]   (time-major so pre[t] slices are
// contiguous [B,4H] blocks)
// ---------------------------------------------------------------------------
__global__ void make_rowidx_kernel(const long long* __restrict__ seq,
                                   int* __restrict__ rowidx) {
  int m = blockIdx.x * blockDim.x + threadIdx.x;
  if (m < B_ * T_) {
    int t = m / B_;
    int b = m % B_;
    rowidx[m] = (int)seq[(long long)b * T_ + t];
  }
}

// ---------------------------------------------------------------------------
// Generic WMMA-f32 GEMM:
//   C[M,N] = act( A[rows[m],K] @ W[N,K]^T + bias1[n] + bias2[n]
//                 + rowadd[m % rowadd_mod, n] )
// Tile: 128(M) x 64(N) per workgroup, 8 waves, each wave = 16 rows x 64 cols.
// K chunked by 32 through LDS. Requires M%128==0, N%64==0, K%32==0.
// ---------------------------------------------------------------------------
__global__ __launch_bounds__(256)
void gemm_wmma_f32_kernel(const float* __restrict__ A,
                          const float* __restrict__ W,
                          const int* __restrict__ a_rows,
                          const float* __restrict__ bias1,
                          const float* __restrict__ bias2,
                          const float* __restrict__ rowadd, int rowadd_mod,
                          float* __restrict__ C,
                          int M, int N, int K, int ldw, int act) {
  __shared__ __align__(16) float a_lds[128 * 32];
  __shared__ __align__(16) float w_lds[64 * 32];

  const int tid  = threadIdx.x;
  const int lane = tid & 31;
  const int wv   = tid >> 5;
  const int n0   = blockIdx.x * 64;
  const int m0   = blockIdx.y * 128;

  v8f acc[4] = {};

  for (int kb = 0; kb < K; kb += 32) {
    // ---- stage A chunk: 128 rows x 32 floats (1024 float4 slots) ----
#pragma unroll
    for (int i = 0; i < 4; ++i) {
      int s   = tid + 256 * i;
      int row = s >> 3, k4 = s & 7;
      int arow = m0 + row;
      int grow = a_rows ? a_rows[arow] : arow;
      float4 v = *(const float4*)(A + (long long)grow * K + kb + 4 * k4);
      *(float4*)(a_lds + row * 32 + 4 * k4) = v;
    }
    // ---- stage W chunk: 64 rows x 32 floats (512 float4 slots) ----
#pragma unroll
    for (int i = 0; i < 2; ++i) {
      int s   = tid + 256 * i;
      int row = s >> 3, k4 = s & 7;
      float4 v = *(const float4*)(W + (long long)(n0 + row) * ldw + kb + 4 * k4);
      *(float4*)(w_lds + row * 32 + 4 * k4) = v;
    }
    // prefetch next weight chunk into cache (global_prefetch_b8)
    if (kb + 32 < K) {
      __builtin_prefetch(W + (long long)(n0 + (tid & 63)) * ldw + kb + 32, 0, 1);
    }
    __syncthreads();

    // ---- 8 K-steps of 4, 4 N-subtiles -> 32 WMMAs per wave per chunk ----
#pragma unroll
    for (int k0 = 0; k0 < 32; k0 += 4) {
      v2f af = *(const v2f*)(a_lds + (16 * wv + (lane & 15)) * 32 + k0 + 2 * (lane >> 4));
#pragma unroll
      for (int s = 0; s < 4; ++s) {
        v2f bf = *(const v2f*)(w_lds + (16 * s + (lane & 15)) * 32 + k0 + 2 * (lane >> 4));
        acc[s] = __builtin_amdgcn_wmma_f32_16x16x4_f32(
            false, af, false, bf, (short)0, acc[s], false, false);
      }
    }
    __syncthreads();
  }

  // ---- epilogue: C layout  VGPR r -> M = r + 8*(lane>=16), N = lane%16 ----
#pragma unroll
  for (int s = 0; s < 4; ++s) {
    int col = n0 + 16 * s + (lane & 15);
    float bv = (bias1 ? bias1[col] : 0.f) + (bias2 ? bias2[col] : 0.f);
#pragma unroll
    for (int r = 0; r < 8; ++r) {
      int row = m0 + 16 * wv + r + 8 * (lane >> 4);
      float v = acc[s][r] + bv;
      if (rowadd) v += rowadd[(long long)(row % rowadd_mod) * N + col];
      if (act == 1) v = fmaxf(v, 0.f);
      C[(long long)row * N + col] = v;
    }
  }
}

// ---------------------------------------------------------------------------
// Fused LSTM step: gates = pre[b,:] (optional) + x@Wx^T (optional) + h@Wh^T
//                  (+ bias1 + bias2), then cell update.
// One workgroup owns batch(128) x 32 hidden columns for ALL FOUR gates, so
// i/f/g/o land in matching lane/VGPR slots and the nonlinearity fuses
// in-registers. Grid = H/32 = 32 workgroups, 256 threads (8 waves).
// All GEMM dims here are K = H = 1024, weights are [4H, H] row-major.
// ---------------------------------------------------------------------------
__global__ __launch_bounds__(256)
void lstm_step_wmma_kernel(const float* __restrict__ pre,   // [B,4H] or null
                           const float* __restrict__ x,     // [B,H] or null
                           const float* __restrict__ Wx,    // [4H,H] or null
                           const float* __restrict__ h,     // [B,H]
                           const float* __restrict__ Wh,    // [4H,H]
                           const float* __restrict__ bias1, // [4H] or null
                           const float* __restrict__ bias2, // [4H] or null
                           float* __restrict__ c,           // [B,H] in/out
                           float* __restrict__ h_out) {     // [B,H] out
  __shared__ __align__(16) float a_lds[128 * 32];
  __shared__ __align__(16) float w_lds[4 * 32 * 32];  // [gate][n][k]

  const int tid  = threadIdx.x;
  const int lane = tid & 31;
  const int wv   = tid >> 5;
  const int n0   = blockIdx.x * 32;

  v8f acc[4][2] = {};  // [gate][n-subtile]

  for (int pass = 0; pass < 2; ++pass) {
    const float* Apt = (pass == 0) ? x : h;
    const float* Wpt = (pass == 0) ? Wx : Wh;
    if (!Apt) continue;  // uniform branch

    for (int kb = 0; kb < H_; kb += 32) {
      // ---- stage activations: 128 x 32 ----
#pragma unroll
      for (int i = 0; i < 4; ++i) {
        int s   = tid + 256 * i;
        int row = s >> 3, k4 = s & 7;
        float4 v = *(const float4*)(Apt + row * H_ + kb + 4 * k4);
        *(float4*)(a_lds + row * 32 + 4 * k4) = v;
      }
      // ---- stage weights: 4 gates x 32 n x 32 k ----
#pragma unroll
      for (int i = 0; i < 4; ++i) {
        int s = tid + 256 * i;
        int g = s >> 8, rem = s & 255;
        int n = rem >> 3, k4 = rem & 7;
        float4 v = *(const float4*)(Wpt + (long long)(g * H_ + n0 + n) * H_ + kb + 4 * k4);
        *(float4*)(w_lds + (g * 32 + n) * 32 + 4 * k4) = v;
      }
      if (kb + 32 < H_) {
        __builtin_prefetch(Wpt + (long long)((tid >> 6) * H_ + n0 + (tid & 31)) * H_ + kb + 32, 0, 1);
      }
      __syncthreads();

#pragma unroll
      for (int k0 = 0; k0 < 32; k0 += 4) {
        v2f af = *(const v2f*)(a_lds + (16 * wv + (lane & 15)) * 32 + k0 + 2 * (lane >> 4));
#pragma unroll
        for (int g = 0; g < 4; ++g) {
#pragma unroll
          for (int s2 = 0; s2 < 2; ++s2) {
            v2f bf = *(const v2f*)(w_lds + (g * 32 + s2 * 16 + (lane & 15)) * 32 + k0 + 2 * (lane >> 4));
            acc[g][s2] = __builtin_amdgcn_wmma_f32_16x16x4_f32(
                false, af, false, bf, (short)0, acc[g][s2], false, false);
          }
        }
      }
      __syncthreads();
    }
  }

  // ---- fused cell update ----
#pragma unroll
  for (int s2 = 0; s2 < 2; ++s2) {
    int hcol = n0 + 16 * s2 + (lane & 15);
#pragma unroll
    for (int r = 0; r < 8; ++r) {
      int b = 16 * wv + r + 8 * (lane >> 4);
      float gi = acc[0][s2][r];
      float gf = acc[1][s2][r];
      float gg = acc[2][s2][r];
      float go = acc[3][s2][r];
      if (pre) {
        const float* p = pre + (long long)b * (4 * H_);
        gi += p[0 * H_ + hcol];
        gf += p[1 * H_ + hcol];
        gg += p[2 * H_ + hcol];
        go += p[3 * H_ + hcol];
      }
      if (bias1) {
        gi += bias1[0 * H_ + hcol]; gf += bias1[1 * H_ + hcol];
        gg += bias1[2 * H_ + hcol]; go += bias1[3 * H_ + hcol];
      }
      if (bias2) {
        gi += bias2[0 * H_ + hcol]; gf += bias2[1 * H_ + hcol];
        gg += bias2[2 * H_ + hcol]; go += bias2[3 * H_ + hcol];
      }
      float i_ = 1.f / (1.f + expf(-gi));
      float f_ = 1.f / (1.f + expf(-gf));
      float g_ = tanhf(gg);
      float o_ = 1.f / (1.f + expf(-go));
      long long idx = (long long)b * H_ + hcol;
      float cn = f_ * c[idx] + i_ * g_;
      c[idx]     = cn;
      h_out[idx] = o_ * tanhf(cn);
    }
  }
}

// ---------------------------------------------------------------------------
// Host orchestration
// ---------------------------------------------------------------------------
extern "C" void kernel_launch(void* const* d_in, const int* in_sizes, int n_in,
                              void* d_out, int out_size, void* d_ws, size_t ws_size,
                              hipStream_t stream) {
  (void)in_sizes; (void)n_in; (void)out_size; (void)ws_size;

  const float*     img_feat = (const float*)d_in[0];
  const long long* seq      = (const long long*)d_in[1];
  const float*     image_W  = (const float*)d_in[2];
  const float*     image_b  = (const float*)d_in[3];
  const float*     emb      = (const float*)d_in[4];
  // d_in[5..8] = attn_W, attn_b, v_W, v_b : mathematically dead (softmax over
  // a singleton axis == 1), intentionally unused.
  const float*     w_ih0    = (const float*)d_in[9];   // [4H, E+H]
  const float*     w_hh0    = (const float*)d_in[10];  // [4H, H]
  const float*     b_ih0    = (const float*)d_in[11];
  const float*     b_hh0    = (const float*)d_in[12];
  const float*     w_ih1    = (const float*)d_in[13];  // [4H, H]
  const float*     w_hh1    = (const float*)d_in[14];  // [4H, H]
  const float*     b_ih1    = (const float*)d_in[15];
  const float*     b_hh1    = (const float*)d_in[16];
  const float*     fc_W     = (const float*)d_in[17];  // [V, H]
  const float*     fc_b     = (const float*)d_in[18];
  float*           out      = (float*)d_out;           // [B, V]

  // ---- workspace layout (floats) ----
  float* ws      = (float*)d_ws;
  float* img_out = ws;  ws += (size_t)B_ * H_;           //  128x1024
  float* Gctx    = ws;  ws += (size_t)B_ * 4 * H_;       //  128x4096
  float* pre     = ws;  ws += (size_t)B_ * T_ * 4 * H_;  // 4096x4096
  float* h0      = ws;  ws += (size_t)2 * B_ * H_;       // ping-pong
  float* h1      = ws;  ws += (size_t)2 * B_ * H_;       // ping-pong
  float* c0      = ws;  ws += (size_t)B_ * H_;
  float* c1      = ws;  ws += (size_t)B_ * H_;
  int*   rowidx  = (int*)ws;                              // B*T ints

  const size_t st = (size_t)B_ * H_ * sizeof(float);

  // zero initial states (c0, c1, h0[0], h1[0])
  hipMemsetAsync(c0, 0, st, stream);
  hipMemsetAsync(c1, 0, st, stream);
  hipMemsetAsync(h0, 0, st, stream);
  hipMemsetAsync(h1, 0, st, stream);

  // time-major gather indices for the embedding rows
  make_rowidx_kernel<<<(B_ * T_ + 255) / 256, 256, 0, stream>>>(seq, rowidx);

  // img_out = relu(img_feat @ image_W^T + image_b)     [128,1024] K=2048
  gemm_wmma_f32_kernel<<<dim3(H_ / 64, B_ / 128), 256, 0, stream>>>(
      img_feat, image_W, nullptr, image_b, nullptr, nullptr, 1,
      img_out, B_, H_, F_, F_, 1);

  // Gctx = img_out @ w_ih0[:, E:]^T                    [128,4096] K=1024
  gemm_wmma_f32_kernel<<<dim3(4 * H_ / 64, B_ / 128), 256, 0, stream>>>(
      img_out, w_ih0 + E_, nullptr, nullptr, nullptr, nullptr, 1,
      Gctx, B_, 4 * H_, H_, E_ + H_, 0);

  // pre[t*B+b] = emb[seq[b,t]] @ w_ih0[:, :E]^T + Gctx[b] + b_ih0 + b_hh0
  gemm_wmma_f32_kernel<<<dim3(4 * H_ / 64, (B_ * T_) / 128), 256, 0, stream>>>(
      emb, w_ih0, rowidx, b_ih0, b_hh0, Gctx, B_,
      pre, B_ * T_, 4 * H_, E_, E_ + H_, 0);

  // ---- sequential scan: 2 fused WMMA+cell kernels per timestep ----
  for (int t = 0; t < T_; ++t) {
    float* h0_in  = h0 + (size_t)(t & 1) * B_ * H_;
    float* h0_out = h0 + (size_t)((t + 1) & 1) * B_ * H_;
    float* h1_in  = h1 + (size_t)(t & 1) * B_ * H_;
    float* h1_out = h1 + (size_t)((t + 1) & 1) * B_ * H_;

    // layer 0: gates = pre[t] + h0 @ w_hh0^T
    lstm_step_wmma_kernel<<<H_ / 32, 256, 0, stream>>>(
        pre + (size_t)t * B_ * 4 * H_, nullptr, nullptr,
        h0_in, w_hh0, nullptr, nullptr, c0, h0_out);

    // layer 1: gates = h0_new @ w_ih1^T + h1 @ w_hh1^T + b_ih1 + b_hh1
    lstm_step_wmma_kernel<<<H_ / 32, 256, 0, stream>>>(
        nullptr, h0_out, w_ih1,
        h1_in, w_hh1, b_ih1, b_hh1, c1, h1_out);
  }

  // out = h1_final @ fc_W^T + fc_b   (T even -> final state in buffer 0)
  gemm_wmma_f32_kernel<<<dim3(V_ / 64, B_ / 128), 256, 0, stream>>>(
      h1, fc_W, nullptr, fc_b, nullptr, nullptr, 1,
      out, B_, V_, H_, H_, 0);
}